// StochasticTwoLayerRGCN_8332236554312
// MI455X (gfx1250) — compile-verified
//
#include <hip/hip_runtime.h>
#include <hip/hip_bf16.h>

// ---------------------------------------------------------------------------
// Two-layer RGCN for MI455X (gfx1250).
//   layer(x, W, b):  mean_r [ (D_dst^-1/2 A_r D_src^-1/2 (x)) W_r + b_r ]
// Dense fp32 WMMA GEMM (v_wmma_f32_16x16x4_f32) per relation with the A-strip
// staged into LDS via GLOBAL_LOAD_ASYNC_TO_LDS_B128 (ASYNCcnt), then edge
// scatter with global f32 atomics into an L2-resident accumulator;
// norm_dst folded into the per-edge contribution, bias/mean fused at the end.
// ---------------------------------------------------------------------------

typedef __attribute__((ext_vector_type(2))) float v2f;
typedef __attribute__((ext_vector_type(8))) float v8f;
typedef int b128_t __attribute__((vector_size(16)));   // 4 x i32 = 128 bits

#define AS1 __attribute__((address_space(1)))
#define AS3 __attribute__((address_space(3)))

#if defined(__has_builtin)
#if __has_builtin(__builtin_amdgcn_global_load_async_to_lds_b128) && \
    __has_builtin(__builtin_amdgcn_s_wait_asynccnt)
#define USE_ASYNC_LDS 1
#endif
#endif

#define RGCN_N     50000
#define RGCN_R     4
#define RGCN_E     500000
#define RGCN_EMB   128
#define RGCN_HID   128
#define RGCN_OUT   64

// ---------------------------------------------------------------------------
// 1) Degree accumulation: deg_out[r][src], deg_in[r][dst] via f32 atomics.
// ---------------------------------------------------------------------------
__global__ void degree_kernel(const int* __restrict__ src,
                              const int* __restrict__ dst,
                              float* __restrict__ deg_out,
                              float* __restrict__ deg_in,
                              int total, int nEdges, int nNodes) {
    int t = blockIdx.x * blockDim.x + threadIdx.x;
    if (t >= total) return;
    int r = t / nEdges;
    atomicAdd(&deg_out[(size_t)r * nNodes + src[t]], 1.0f);
    atomicAdd(&deg_in [(size_t)r * nNodes + dst[t]], 1.0f);
}

// ---------------------------------------------------------------------------
// 2) In-place norm: v -> (v > 0) ? rsqrt(max(v,1)) : 0
// ---------------------------------------------------------------------------
__global__ void norm_kernel(float* __restrict__ buf, int total) {
    int t = blockIdx.x * blockDim.x + threadIdx.x;
    if (t >= total) return;
    float v = buf[t];
    buf[t] = (v > 0.0f) ? rsqrtf(fmaxf(v, 1.0f)) : 0.0f;
}

// ---------------------------------------------------------------------------
// 3) WMMA GEMM:  Y[N][HOUT] = (X[N][HIN] * ns[:,None]) @ W[HIN][HOUT]
//    blockIdx.x = 16-row strip; wave-in-block = 16-col tile.
//    The 16xHIN A-strip (contiguous 8 KB) is staged into LDS once per block
//    with async copies, then all HOUT/16 waves read A from LDS (ds_load_b64).
//    K-loop in steps of 4 using V_WMMA_F32_16X16X4_F32 (fp32 accumulate).
//
//    Operand layouts per CDNA5 ISA 7.12.2:
//      A 16x4 : lanes 0-15 -> M=0..15; VGPR0={K0,K2}, VGPR1={K1,K3}
//      B 4x16 : lanes 0-15 -> N=0..15; VGPR0={K0,K2}, VGPR1={K1,K3}
//      C/D    : 8 VGPRs; VGPR i: lanes 0-15 -> M=i, lanes 16-31 -> M=8+i
//
//    LDS rows padded to HIN+4 floats: at fixed k the 16 row-readers hit
//    banks 4*row + k -> conflict-free b64 reads; 8 B alignment preserved.
// ---------------------------------------------------------------------------
template <int HIN, int HOUT>
__global__ void gemm_norm_wmma(const float* __restrict__ X,
                               const float* __restrict__ ns,
                               const float* __restrict__ W,
                               float* __restrict__ Y) {
    constexpr int HINP = HIN + 4;               // padded LDS row stride
    __shared__ float tileX[16 * HINP];

    const int lane    = threadIdx.x & 31;
    const int waveId  = threadIdx.x >> 5;       // column tile within block
    const int rowTile = blockIdx.x;             // 16-row strip
    const int half    = lane >> 4;              // selects K pair within step
    const int l15     = lane & 15;

    // ---- stage A-strip (16 rows x HIN floats, contiguous) into LDS ----
    {
        const float* strip = X + (size_t)rowTile * 16 * HIN;
        constexpr int CPR = HIN / 4;            // float4 chunks per row
        constexpr int CH  = 16 * CPR;           // total float4 chunks
        for (int c = threadIdx.x; c < CH; c += blockDim.x) {
            const int row = c / CPR;
            const int col = c - row * CPR;
            float* g = const_cast<float*>(strip + row * HIN + col * 4);
            float* l = &tileX[row * HINP + col * 4];
#ifdef USE_ASYNC_LDS
            __builtin_amdgcn_global_load_async_to_lds_b128(
                (AS1 b128_t*)g, (AS3 b128_t*)l, /*offset=*/0, /*cpol=*/0);
#else
            const float4 v = *reinterpret_cast<const float4*>(g);
            *reinterpret_cast<float4*>(l) = v;
#endif
        }
    }
#ifdef USE_ASYNC_LDS
    __builtin_amdgcn_s_wait_asynccnt(0);
#endif
    __syncthreads();

    const int m = rowTile * 16 + l15;           // A row handled by this lane
    const int n = waveId * 16 + l15;            // B/D column handled by this lane

    const float s = ns[m];                      // norm_src scale for this row
    const float* __restrict__ wcol = W + n;

    v8f c = {};
    for (int k0 = 0; k0 < HIN; k0 += 4) {
        const int kk = k0 + half * 2;
        // A from LDS (ds_load_b64), scaled by norm_src
        v2f a = *reinterpret_cast<const v2f*>(&tileX[l15 * HINP + kk]);
        a.x *= s;
        a.y *= s;
        // B streamed from global/L2
        v2f b;
        b.x = wcol[(size_t)kk       * HOUT];
        b.y = wcol[(size_t)(kk + 1) * HOUT];
        // 8 args: (neg_a, A, neg_b, B, c_mod, C, reuse_a, reuse_b)
        c = __builtin_amdgcn_wmma_f32_16x16x4_f32(false, a, false, b,
                                                  (short)0, c, false, false);
    }

    // Store D: VGPR i holds row (half*8 + i) of the tile, column l15.
    float* __restrict__ yrow =
        Y + (size_t)(rowTile * 16 + half * 8) * HOUT + waveId * 16 + l15;
#pragma unroll
    for (int i = 0; i < 8; ++i)
        yrow[(size_t)i * HOUT] = c[i];
}

// ---------------------------------------------------------------------------
// 4) Edge scatter: one wave per edge; lane handles H/32 contiguous floats.
//    acc[dst] += xs[src] * norm_dst[dst]   (global f32 atomics, L2-resident)
// ---------------------------------------------------------------------------
template <int H>
__global__ void scatter_kernel(const float* __restrict__ xs,
                               const int* __restrict__ src,
                               const int* __restrict__ dst,
                               const float* __restrict__ nd,
                               float* __restrict__ acc,
                               int nEdges) {
    const int edge = (int)((blockIdx.x * blockDim.x + threadIdx.x) >> 5);
    const int lane = threadIdx.x & 31;
    if (edge >= nEdges) return;

    const int s = src[edge];
    const int d = dst[edge];
    const float w = nd[d];

    constexpr int V = H / 32;
    const float* __restrict__ xrow = xs + (size_t)s * H + lane * V;
    float* __restrict__ arow       = acc + (size_t)d * H + lane * V;

    if constexpr (V == 4) {
        const float4 v = *reinterpret_cast<const float4*>(xrow);
        atomicAdd(&arow[0], v.x * w);
        atomicAdd(&arow[1], v.y * w);
        atomicAdd(&arow[2], v.z * w);
        atomicAdd(&arow[3], v.w * w);
    } else {
        const float2 v = *reinterpret_cast<const float2*>(xrow);
        atomicAdd(&arow[0], v.x * w);
        atomicAdd(&arow[1], v.y * w);
    }
}

// ---------------------------------------------------------------------------
// 5) Finalize: out = 0.25 * (acc + sum_r b[r])        (mean over 4 relations)
// ---------------------------------------------------------------------------
__global__ void finalize_kernel(const float* __restrict__ acc,
                                const float* __restrict__ b,   // [4][H]
                                float* __restrict__ out,
                                int total, int H) {
    int t = blockIdx.x * blockDim.x + threadIdx.x;
    if (t >= total) return;
    int j = t % H;
    float bias = b[j] + b[H + j] + b[2 * H + j] + b[3 * H + j];
    out[t] = 0.25f * (acc[t] + bias);
}

// ---------------------------------------------------------------------------
// Launch
// ---------------------------------------------------------------------------
extern "C" void kernel_launch(void* const* d_in, const int* in_sizes, int n_in,
                              void* d_out, int out_size, void* d_ws, size_t ws_size,
                              hipStream_t stream) {
    (void)in_sizes; (void)n_in; (void)out_size; (void)ws_size;

    const int N = RGCN_N, R = RGCN_R, E = RGCN_E;
    const int HIN = RGCN_EMB, HID = RGCN_HID, HOUT = RGCN_OUT;

    // setup_inputs order: node_ids, src, dst, emb, W1, b1, W2, b2
    const int*   src = (const int*)  d_in[1];   // [R][E]
    const int*   dst = (const int*)  d_in[2];   // [R][E]
    const float* emb = (const float*)d_in[3];   // [N][128]
    const float* W1  = (const float*)d_in[4];   // [R][128][128]
    const float* b1  = (const float*)d_in[5];   // [R][128]
    const float* W2  = (const float*)d_in[6];   // [R][128][64]
    const float* b2  = (const float*)d_in[7];   // [R][64]
    float*       out = (float*)d_out;           // [N][64]

    // Workspace layout (floats)
    float* ws  = (float*)d_ws;
    float* ns  = ws;                      // [R][N] deg_out -> norm_src (in place)
    float* nd  = ns  + (size_t)R * N;     // [R][N] deg_in  -> norm_dst (in place)
    float* xs  = nd  + (size_t)R * N;     // [N][128] per-relation projection
    float* acc = xs  + (size_t)N * HID;   // [N][128] scatter accumulator
    float* h1  = acc + (size_t)N * HID;   // [N][128] layer-1 output

    // --- degrees + norms (shared by both layers) ---
    (void)hipMemsetAsync(ns, 0, (size_t)2 * R * N * sizeof(float), stream);
    {
        int total = R * E;
        degree_kernel<<<(total + 255) / 256, 256, 0, stream>>>(src, dst, ns, nd,
                                                               total, E, N);
        int tn = 2 * R * N;
        norm_kernel<<<(tn + 255) / 256, 256, 0, stream>>>(ns, tn);
    }

    const int rowTiles = N / 16;  // 3125, exact

    // --- layer 1: 128 -> 128 ---
    (void)hipMemsetAsync(acc, 0, (size_t)N * HID * sizeof(float), stream);
    for (int r = 0; r < R; ++r) {
        gemm_norm_wmma<RGCN_EMB, RGCN_HID><<<rowTiles, 32 * (HID / 16), 0, stream>>>(
            emb, ns + (size_t)r * N, W1 + (size_t)r * HIN * HID, xs);
        scatter_kernel<RGCN_HID><<<(E * 32 + 255) / 256, 256, 0, stream>>>(
            xs, src + (size_t)r * E, dst + (size_t)r * E, nd + (size_t)r * N,
            acc, E);
    }
    {
        int total = N * HID;
        finalize_kernel<<<(total + 255) / 256, 256, 0, stream>>>(acc, b1, h1,
                                                                 total, HID);
    }

    // --- layer 2: 128 -> 64 ---
    (void)hipMemsetAsync(acc, 0, (size_t)N * HOUT * sizeof(float), stream);
    for (int r = 0; r < R; ++r) {
        gemm_norm_wmma<RGCN_HID, RGCN_OUT><<<rowTiles, 32 * (HOUT / 16), 0, stream>>>(
            h1, ns + (size_t)r * N, W2 + (size_t)r * HID * HOUT, xs);
        scatter_kernel<RGCN_OUT><<<(E * 32 + 255) / 256, 256, 0, stream>>>(
            xs, src + (size_t)r * E, dst + (size_t)r * E, nd + (size_t)r * N,
            acc, E);
    }
    {
        int total = N * HOUT;
        finalize_kernel<<<(total + 255) / 256, 256, 0, stream>>>(acc, b2, out,
                                                                 total, HOUT);
    }
}